// Kernel_21638045237484
// MI455X (gfx1250) — compile-verified
//
#include <hip/hip_runtime.h>
#include <cmath>

// ---------------------------------------------------------------------------
// e3nn-style tensor-product kernel builder for MI455X (gfx1250, wave32).
//
// out[z, row, col], z in [0,1024), 288x288 block matrix per z.
// Block (lo,li): E[uv(1024), ab] = c_z(1024 x nlf) @ D(nlf x ab), per-z GEMM,
// computed with V_WMMA_F32_16X16X4_F32 (f32 exact vs reference).
//
// Roofline: ~420 MB traffic (340 MB stores) vs 0.54 GFLOP -> store-bound,
// ~18 us floor at 23.3 TB/s. All operands padded in LDS so the WMMA inner
// loop is branch-free: ds_load_b64 + v_wmma per K-step (B hoisted).
// ---------------------------------------------------------------------------

typedef float v2f __attribute__((ext_vector_type(2)));
typedef float v8f __attribute__((ext_vector_type(8)));

#define NBATCH   1024
#define NPATH    19456
#define OUTDIM   288
#define NCOMBO   19
#define D_TOTAL  259
#define CG_TOTAL 1225

// Per-combo tables (compile-time foldable).
constexpr int k_lo[NCOMBO]    = {0,0,0,1,1,1,1,1,1,1,2,2,2,2,2,2,2,2,2};
constexpr int k_li[NCOMBO]    = {0,1,2,0,1,1,1,2,2,2,0,1,1,1,2,2,2,2,2};
constexpr int k_lf[NCOMBO]    = {0,1,2,1,0,1,2,1,2,3,2,1,2,3,0,1,2,3,4};
constexpr int k_cgoff[NCOMBO] = {0,1,10,35,44,53,80,125,170,245,350,375,420,495,600,625,700,825,1000};
constexpr int k_doff[NCOMBO]  = {0,1,4,9,12,21,30,39,54,69,84,89,104,119,134,159,184,209,234};

// ---------------------------------------------------------------------------
// Device CG computation (f64): real-basis Clebsch-Gordan via complex q-basis.
// Runs once on one workgroup; cost irrelevant.
// ---------------------------------------------------------------------------
__device__ double dfact(int n) {
    double r = 1.0;
    for (int i = 2; i <= n; ++i) r *= (double)i;
    return r;
}

__device__ double su2cg(int j1, int m1, int j2, int m2, int j3, int m3) {
    if (m1 + m2 != m3 || j3 < abs(j1 - j2) || j3 > j1 + j2) return 0.0;
    double pref = sqrt((2.0 * j3 + 1.0) * dfact(j3 + j1 - j2) * dfact(j3 - j1 + j2) *
                       dfact(j1 + j2 - j3) / dfact(j1 + j2 + j3 + 1));
    pref *= sqrt(dfact(j3 + m3) * dfact(j3 - m3) * dfact(j1 - m1) * dfact(j1 + m1) *
                 dfact(j2 - m2) * dfact(j2 + m2));
    double s = 0.0;
    for (int v = 0; v <= j1 + j2 - j3; ++v) {
        int a0 = v, a1 = j1 + j2 - j3 - v, a2 = j1 - m1 - v;
        int a3 = j2 + m2 - v, a4 = j3 - j2 + m1 + v, a5 = j3 - j1 - m2 + v;
        if (a0 < 0 || a1 < 0 || a2 < 0 || a3 < 0 || a4 < 0 || a5 < 0) continue;
        double d = dfact(a0) * dfact(a1) * dfact(a2) * dfact(a3) * dfact(a4) * dfact(a5);
        s += ((v & 1) ? -1.0 : 1.0) / d;
    }
    return pref * s;
}

// q(l)[r][c] including the (-i)^l prefactor.
__device__ void q_elem(int l, int r, int c, double& re, double& im) {
    int m = r - l;
    double is2 = 0.7071067811865475244;
    double br = 0.0, bi = 0.0;
    if (m < 0) {
        if (c == l - m)      br = is2;    // col l+|m| : 1/sqrt2
        else if (c == l + m) bi = -is2;   // col l-|m| : -i/sqrt2
    } else if (m == 0) {
        if (c == l) br = 1.0;
    } else {
        double s = (m & 1) ? -1.0 : 1.0;
        if (c == l + m)      br = s * is2;
        else if (c == l - m) bi = s * is2; // i*(-1)^m/sqrt2
    }
    switch (l & 3) {  // multiply by (-i)^l
        case 0: re =  br; im =  bi; break;
        case 1: re =  bi; im = -br; break;
        case 2: re = -br; im = -bi; break;
        default: re = -bi; im =  br; break;
    }
}

__global__ void cg_kernel(float* __restrict__ cgbuf) {
    int t = threadIdx.x;
    for (int c = 0; c < NCOMBO; ++c) {
        int l1 = k_lo[c], l2 = k_li[c], l3 = k_lf[c];
        int w1 = 2 * l1 + 1, w2 = 2 * l2 + 1, w3 = 2 * l3 + 1;
        int sz = w1 * w2 * w3;
        for (int e = t; e < sz; e += blockDim.x) {
            int j = e / (w2 * w3);
            int rem = e % (w2 * w3);
            int lcol = rem / w3;
            int m = rem % w3;
            double s = 0.0;
            for (int i = 0; i < w1; ++i) {
                int m1 = i - l1;
                double q1r, q1i;
                q_elem(l1, i, j, q1r, q1i);
                if (q1r == 0.0 && q1i == 0.0) continue;
                for (int k = 0; k < w2; ++k) {
                    int m2 = k - l2;
                    int m3 = m1 + m2;
                    if (m3 < -l3 || m3 > l3) continue;
                    double q2r, q2i;
                    q_elem(l2, k, lcol, q2r, q2i);
                    if (q2r == 0.0 && q2i == 0.0) continue;
                    double q3r, q3i;
                    q_elem(l3, l3 + m3, m, q3r, q3i);
                    if (q3r == 0.0 && q3i == 0.0) continue;
                    double su = su2cg(l1, m1, l2, m2, l3, m3);
                    double pr = q1r * q2r - q1i * q2i;
                    double pi = q1r * q2i + q1i * q2r;
                    s += su * (pr * q3r + pi * q3i);   // Re((q1 q2) * conj(q3))
                }
            }
            cgbuf[k_cgoff[c] + e] = (float)s;
        }
    }
}

// ---------------------------------------------------------------------------
// One fully-specialized block GEMM: E = c_z @ D via f32 WMMA 16x16x4.
// All shape parameters are compile-time; LDS operands are zero-padded inline
// during the fill pass so the inner loop has no conditionals (EXEC stays
// all-ones around v_wmma).
// ---------------------------------------------------------------------------
template <int NLF, int LO, int LI, int PO, int DBASE, int RBASE, int CBASE>
__device__ __forceinline__ void do_block(const float* __restrict__ cz,
                                         const float* __restrict__ sD,
                                         float* __restrict__ sC,
                                         float* __restrict__ sB,
                                         float* __restrict__ out,
                                         size_t outZ, int tid) {
    constexpr int NROWS  = 2 * LO + 1;
    constexpr int NCOLS  = 2 * LI + 1;
    constexpr int NCOUNT = NROWS * NCOLS;        // N of the GEMM (<=25)
    constexpr int KSTEPS = (NLF + 3) / 4;
    constexpr int KP     = 4 * KSTEPS;           // padded K (4 or 8)
    constexpr int NTILES = (NCOUNT + 15) / 16;
    constexpr int NP     = 16 * NTILES;          // padded N

    // Single-pass staging with inline zero padding.
    // sC: padded [M][KP] coefficient layout (coalesced-ish global read).
    for (int t = tid; t < 1024 * KP; t += 256) {
        int m = t / KP, lf = t % KP;
        sC[t] = (lf < NLF) ? cz[PO + m * NLF + lf] : 0.0f;
    }
    // sB: padded, transposed B tile: sB[n*KP + k] = D[k][n].
    for (int t = tid; t < NP * KP; t += 256) {
        int n = t / KP, k = t % KP;
        sB[t] = (k < NLF && n < NCOUNT) ? sD[DBASE + k * NCOUNT + n] : 0.0f;
    }
    __syncthreads();

    const int lane   = tid & 31;
    const int wid    = tid >> 5;
    const int laneHi = lane >> 4;     // 0 for lanes 0-15, 1 for 16-31
    const int nl     = lane & 15;

    for (int mt = wid; mt < 64; mt += 8) {
        const int Mrow = mt * 16 + nl;                    // A row = lane%16
        const float* aptr = sC + Mrow * KP + 2 * laneHi;  // K = 2*laneHi + g
#pragma unroll
        for (int nt = 0; nt < NTILES; ++nt) {
            const int ng = nt * 16 + nl;                  // B column for this lane
            const float* bptr = sB + ng * KP + 2 * laneHi;
            v8f acc = {};
#pragma unroll
            for (int ks = 0; ks < KSTEPS; ++ks) {
                v2f av = *(const v2f*)(aptr + 4 * ks);    // ds_load_b64
                v2f bv = *(const v2f*)(bptr + 4 * ks);    // ds_load_b64 (hoisted)
                acc = __builtin_amdgcn_wmma_f32_16x16x4_f32(
                    false, av, false, bv, (short)0, acc, false, false);
            }
            if (ng < NCOUNT) {
                const int a  = ng / NCOLS;
                const int bb = ng % NCOLS;
#pragma unroll
                for (int g = 0; g < 8; ++g) {
                    // C/D layout: VGPR g -> M = g (lanes 0-15), g+8 (lanes 16-31)
                    const int M = mt * 16 + g + laneHi * 8;
                    const int u = M >> 5, v = M & 31;
                    const int row = RBASE + u * NROWS + a;
                    const int col = CBASE + v * NCOLS + bb;
                    out[outZ + (size_t)row * OUTDIM + col] = acc[g];
                }
            }
        }
    }
    __syncthreads();
}

// ---------------------------------------------------------------------------
// Main kernel: one workgroup per batch element z, 8 waves.
// ---------------------------------------------------------------------------
__global__ __launch_bounds__(256) void tp_kernel(const float* __restrict__ Ys,
                                                 const float* __restrict__ coeff,
                                                 const float* __restrict__ nrm,
                                                 const float* __restrict__ cg,
                                                 float* __restrict__ out) {
    __shared__ alignas(16) float sY[32];
    __shared__ alignas(16) float sD[D_TOTAL];
    __shared__ alignas(16) float sB[8 * 32];    // max KP*NP = 8*32
    __shared__ alignas(16) float sC[1024 * 8];  // max 1024*KP

    const int z   = blockIdx.x;
    const int tid = threadIdx.x;
    const float* cz = coeff + (size_t)z * NPATH;

    // Warm the cache with this z's whole coefficient row (re-read per block).
    for (int t = tid * 32; t < NPATH; t += 256 * 32)
        __builtin_prefetch(cz + t, 0, 1);       // global_prefetch_b8

    if (tid < 25) sY[tid] = Ys[tid * NBATCH + z];
    __syncthreads();

    // D_lf[a,b] = nrm * sum_k CG[a,b,k] * Y[lf^2 + k, z]   (259 floats)
#pragma unroll
    for (int c = 0; c < NCOMBO; ++c) {
        const int lo = k_lo[c], li = k_li[c], lf = k_lf[c];
        const int wi = 2 * li + 1, wo = 2 * lo + 1, wf = 2 * lf + 1;
        const int nab = wo * wi;
        if (tid < nab) {
            const float* C = cg + k_cgoff[c] + tid * wf;
            const float* y = sY + lf * lf;
            float s = 0.0f;
            for (int k = 0; k < wf; ++k) s += C[k] * y[k];
            float n = nrm[(lo * 3 + li) * NBATCH + z];
            sD[k_doff[c] + tid] = s * n;
        }
    }
    __syncthreads();

    const size_t outZ = (size_t)z * (OUTDIM * OUTDIM);

    //       NLF LO LI    PO  DBASE RBASE CBASE
    do_block<1, 0, 0,     0,   0,   0,   0>(cz, sD, sC, sB, out, outZ, tid);
    do_block<1, 0, 1,  1024,   1,   0,  32>(cz, sD, sC, sB, out, outZ, tid);
    do_block<1, 0, 2,  2048,   4,   0, 128>(cz, sD, sC, sB, out, outZ, tid);
    do_block<1, 1, 0,  3072,   9,  32,   0>(cz, sD, sC, sB, out, outZ, tid);
    do_block<3, 1, 1,  4096,  12,  32,  32>(cz, sD, sC, sB, out, outZ, tid);
    do_block<3, 1, 2,  7168,  39,  32, 128>(cz, sD, sC, sB, out, outZ, tid);
    do_block<1, 2, 0, 10240,  84, 128,   0>(cz, sD, sC, sB, out, outZ, tid);
    do_block<3, 2, 1, 11264,  89, 128,  32>(cz, sD, sC, sB, out, outZ, tid);
    do_block<5, 2, 2, 14336, 134, 128, 128>(cz, sD, sC, sB, out, outZ, tid);
}

extern "C" void kernel_launch(void* const* d_in, const int* in_sizes, int n_in,
                              void* d_out, int out_size, void* d_ws, size_t ws_size,
                              hipStream_t stream) {
    const float* Ys    = (const float*)d_in[0];  // (25, 1024)
    const float* coeff = (const float*)d_in[1];  // (1024, 19456)
    const float* nrm   = (const float*)d_in[2];  // (3, 3, 1024)
    float* out         = (float*)d_out;          // (1024, 288, 288)
    float* cgbuf       = (float*)d_ws;           // 1225 floats of scratch

    cg_kernel<<<dim3(1), dim3(256), 0, stream>>>(cgbuf);
    tp_kernel<<<dim3(NBATCH), dim3(256), 0, stream>>>(Ys, coeff, nrm, cgbuf, out);
    (void)in_sizes; (void)n_in; (void)out_size; (void)ws_size;
}